// SOMLayer_42631845380411
// MI455X (gfx1250) — compile-verified
//
#include <hip/hip_runtime.h>

typedef float v2f __attribute__((ext_vector_type(2)));
typedef float v8f __attribute__((ext_vector_type(8)));

#define SOM_D       256
#define SOM_N       2048
#define ROWS_PER_WG 32     // two 16-row M-tiles per wave -> halved codebook traffic
#define ZT_STRIDE   260    // 256 + 4 pad -> conflict-free A-fragment ds reads
// ALPHA_SOM = 5.0 ; logit = -(alpha+1)/2 * log1p(d/alpha) = -3 * log1p(d * 0.2)

__device__ inline v8f wmma4(v2f a, v2f b, v8f c) {
  return __builtin_amdgcn_wmma_f32_16x16x4_f32(
      /*neg_a=*/false, a, /*neg_b=*/false, b,
      /*c_mod=*/(short)0, c, /*reuse_a=*/false, /*reuse_b=*/false);
}

// ---------------------------------------------------------------------------
// Kernel 0: codebook squared norms -> workspace (2048 floats)
// ---------------------------------------------------------------------------
__global__ __launch_bounds__(256) void som_cnorm_kernel(
    const float* __restrict__ cb, float* __restrict__ cnorm) {
  const int wid  = threadIdx.x >> 5;
  const int lane = threadIdx.x & 31;
  const int row  = blockIdx.x * 8 + wid;
  const float* p = cb + (size_t)row * SOM_D;
  float s = 0.0f;
#pragma unroll
  for (int c = 0; c < SOM_D / 32; ++c) {
    float v = p[lane + c * 32];
    s = fmaf(v, v, s);
  }
#pragma unroll
  for (int off = 16; off; off >>= 1) s += __shfl_xor(s, off, 32);
  if (lane == 0) cnorm[row] = s;
}

// ---------------------------------------------------------------------------
// Kernel 1: fused distance-GEMM (f32 WMMA, 2 M-tiles/wave) + argmin + softmax.
// Block = 512 threads = 16 wave32. Block owns 32 query rows x all 2048 cols.
// Wave w owns columns [w*128, w*128+128) for BOTH 16-row M-tiles, so each
// B fragment (codebook) load feeds two v_wmma.
// Softmax/argmin are register-resident: partials from the WMMA accumulators,
// combined via small LDS arrays; q/q_ng streamed out with NT stores.
// ---------------------------------------------------------------------------
__global__ __launch_bounds__(512) void som_main_kernel(
    const float* __restrict__ z, const float* __restrict__ cb,
    const float* __restrict__ cnorm, float* __restrict__ q,
    float* __restrict__ qng, float* __restrict__ bmu,
    float* __restrict__ zq) {
  __shared__ float zt[ROWS_PER_WG * ZT_STRIDE];   // ~33 KB
  __shared__ float znorm_s[ROWS_PER_WG];
  __shared__ float pmin_d[ROWS_PER_WG * 16];      // per-(row, wave) partial min
  __shared__ int   pmin_i[ROWS_PER_WG * 16];
  __shared__ float psum_s[ROWS_PER_WG * 16];      // per-(row, wave) partial sum
  __shared__ float rowmax_s[ROWS_PER_WG];         // row max-logit
  __shared__ float rowinv_s[ROWS_PER_WG];         // 1 / row sum
  __shared__ int   bmu_s[ROWS_PER_WG];

  const int tid  = threadIdx.x;
  const int wid  = tid >> 5;
  const int lane = tid & 31;
  const int mrow = lane & 15;   // column-within-tile / reducer id
  const int half = lane >> 4;   // WMMA half-wave select
  const int m0   = blockIdx.x * ROWS_PER_WG;

  // ---- Phase A: cooperative load of the z tile (32 x 256) ----
  {
    const int row = tid >> 4;          // 0..31
    const int c0  = (tid & 15) * 16;   // 16 threads x 16 floats = 256
    const float4* src = (const float4*)(z + (size_t)(m0 + row) * SOM_D + c0);
    float4* dst = (float4*)(&zt[row * ZT_STRIDE + c0]);
    dst[0] = src[0]; dst[1] = src[1]; dst[2] = src[2]; dst[3] = src[3];
  }
  __syncthreads();

  // ---- per-row ||z||^2 : wave w reduces rows w and w+16 ----
#pragma unroll
  for (int t = 0; t < 2; ++t) {
    const int row = wid + 16 * t;
    float s = 0.0f;
#pragma unroll
    for (int c = 0; c < SOM_D / 32; ++c) {
      float v = zt[row * ZT_STRIDE + lane + c * 32];
      s = fmaf(v, v, s);
    }
#pragma unroll
    for (int off = 16; off; off >>= 1) s += __shfl_xor(s, off, 32);
    if (lane == 0) znorm_s[row] = s;
  }
  __syncthreads();

  // ---- Phase B: GEMM via V_WMMA_F32_16X16X4_F32, register-blocked in M ----
  const int n0 = wid * 128;
  const float* bbase = cb + (size_t)(n0 + mrow) * SOM_D + 2 * half;
  const float* a0b = &zt[mrow * ZT_STRIDE + 2 * half];
  const float* a1b = &zt[(16 + mrow) * ZT_STRIDE + 2 * half];

  v8f acc0[8] = {};
  v8f acc1[8] = {};

#pragma unroll 2
  for (int k0 = 0; k0 < SOM_D; k0 += 4) {
    v2f a0 = *(const v2f*)(a0b + k0);
    v2f a1 = *(const v2f*)(a1b + k0);
#pragma unroll
    for (int j = 0; j < 8; ++j) {
      v2f b = *(const v2f*)(bbase + (size_t)j * 16 * SOM_D + k0);
      acc0[j] = wmma4(a0, b, acc0[j]);   // B reused for both M-tiles
      acc1[j] = wmma4(a1, b, acc1[j]);
    }
  }

  // per-lane codebook norms for this wave's 8 column groups
  float cn[8];
#pragma unroll
  for (int j = 0; j < 8; ++j) cn[j] = cnorm[n0 + j * 16 + mrow];

  // ---- Phase C1: per-wave partial argmin from registers ----
  auto partial_min = [&](v8f (&acc)[8], int t) {
#pragma unroll
    for (int r = 0; r < 8; ++r) {
      const int row = t * 16 + r + 8 * half;
      const float zn = znorm_s[row];
      float mind = 3.402823466e38f;
      int mini = 0;
#pragma unroll
      for (int j = 0; j < 8; ++j) {  // cols ascend with j -> strict < keeps 1st
        float dd = fmaxf(zn + cn[j] - 2.0f * acc[j][r], 0.0f);
        if (dd < mind) { mind = dd; mini = n0 + j * 16 + mrow; }
      }
#pragma unroll
      for (int off = 8; off; off >>= 1) {  // reduce within 16-lane group
        float od = __shfl_xor(mind, off, 32);
        int   oi = __shfl_xor(mini, off, 32);
        if (od < mind || (od == mind && oi < mini)) { mind = od; mini = oi; }
      }
      if (mrow == 0) { pmin_d[row * 16 + wid] = mind; pmin_i[row * 16 + wid] = mini; }
    }
  };
  partial_min(acc0, 0);
  partial_min(acc1, 1);
  __syncthreads();

  // ---- combine mins: lanes 0-15 -> row wid, lanes 16-31 -> row wid+16 ----
  {
    const int row = wid + 16 * half;
    float mind = pmin_d[row * 16 + mrow];
    int   mini = pmin_i[row * 16 + mrow];
#pragma unroll
    for (int off = 8; off; off >>= 1) {
      float od = __shfl_xor(mind, off, 32);
      int   oi = __shfl_xor(mini, off, 32);
      if (od < mind || (od == mind && oi < mini)) { mind = od; mini = oi; }
    }
    if (mrow == 0) {
      // logit monotone-decreasing in dist => row max-logit at min-dist
      rowmax_s[row] = -3.0f * __logf(fmaf(mind, 0.2f, 1.0f));
      bmu_s[row] = mini;
      bmu[m0 + row] = (float)mini;
    }
  }
  __syncthreads();

  // ---- Phase C2: exp + per-wave partial row sums (exp cached into acc) ----
  auto partial_sum = [&](v8f (&acc)[8], int t) {
#pragma unroll
    for (int r = 0; r < 8; ++r) {
      const int row = t * 16 + r + 8 * half;
      const float zn = znorm_s[row];
      const float rm = rowmax_s[row];
      float ps = 0.0f;
#pragma unroll
      for (int j = 0; j < 8; ++j) {
        float dd = fmaxf(zn + cn[j] - 2.0f * acc[j][r], 0.0f);
        float e  = __expf(fmaf(-3.0f, __logf(fmaf(dd, 0.2f, 1.0f)), -rm));
        acc[j][r] = e;
        ps += e;
      }
#pragma unroll
      for (int off = 8; off; off >>= 1) ps += __shfl_xor(ps, off, 32);
      if (mrow == 0) psum_s[row * 16 + wid] = ps;
    }
  };
  partial_sum(acc0, 0);
  partial_sum(acc1, 1);
  __syncthreads();

  {
    const int row = wid + 16 * half;
    float s = psum_s[row * 16 + mrow];
#pragma unroll
    for (int off = 8; off; off >>= 1) s += __shfl_xor(s, off, 32);
    if (mrow == 0) rowinv_s[row] = 1.0f / s;
  }
  __syncthreads();

  // ---- Phase C3: scale + NT-store q and q_ng straight from registers ----
  auto store_q = [&](v8f (&acc)[8], int t) {
#pragma unroll
    for (int r = 0; r < 8; ++r) {
      const int row = t * 16 + r + 8 * half;
      const float inv = rowinv_s[row];
      const size_t base = (size_t)(m0 + row) * SOM_N + n0 + mrow;
#pragma unroll
      for (int j = 0; j < 8; ++j) {
        float qv = acc[j][r] * inv;
        __builtin_nontemporal_store(qv, q + base + j * 16);     // q
        __builtin_nontemporal_store(qv, qng + base + j * 16);   // q_ng == q fwd
      }
    }
  };
  store_q(acc0, 0);
  store_q(acc1, 1);

  // ---- z_q gather: wave w copies rows w and w+16 ----
#pragma unroll
  for (int t = 0; t < 2; ++t) {
    const int row = wid + 16 * t;
    const int mini = bmu_s[row];
    const float* crow = cb + (size_t)mini * SOM_D;
    float* zr = zq + (size_t)(m0 + row) * SOM_D;
#pragma unroll
    for (int c = 0; c < SOM_D / 32; ++c)
      zr[lane + c * 32] = crow[lane + c * 32];
  }
}

// ---------------------------------------------------------------------------
extern "C" void kernel_launch(void* const* d_in, const int* in_sizes, int n_in,
                              void* d_out, int out_size, void* d_ws,
                              size_t ws_size, hipStream_t stream) {
  const float* z  = (const float*)d_in[0];   // [B*T, 256] f32
  const float* cb = (const float*)d_in[1];   // [2048, 256] f32
  const int M = in_sizes[0] / SOM_D;         // 16384

  float* cnorm = (float*)d_ws;               // 2048 floats of scratch
  float* out   = (float*)d_out;
  const size_t qElems = (size_t)M * SOM_N;
  float* q   = out;
  float* qng = out + qElems;
  float* bmu = out + 2 * qElems;
  float* zq  = out + 2 * qElems + M;

  som_cnorm_kernel<<<SOM_N / 8, 256, 0, stream>>>(cb, cnorm);
  som_main_kernel<<<M / ROWS_PER_WG, 512, 0, stream>>>(z, cb, cnorm, q, qng,
                                                       bmu, zq);
}